// FrameCanonicalProjection_59957743452495
// MI455X (gfx1250) — compile-verified
//
#include <hip/hip_runtime.h>

typedef __attribute__((ext_vector_type(2))) float v2f;
typedef __attribute__((ext_vector_type(4))) float v4f;
typedef __attribute__((ext_vector_type(8))) float v8f;

#define NUM_FRAMES 4
#define RD 14            // relation dim
#define ND 64            // canonical dim
#define ROW_PAD 68       // padded A-stage row stride (words) -> bank-conflict-free frag reads
#define WAVES 8
#define ROWS_PER_WAVE 16
#define ROWS_PER_BLOCK (WAVES * ROWS_PER_WAVE)   // 128

// out = A_aug @ W_aug, where augmented K=64 encodes expert routing + bias:
//   k in [14e, 14e+14) : x-slot of expert e      (A has x there iff id==e)
//   k == 56+e          : bias one-hot for expert e
//   k in [60,64)       : zero padding
__global__ __launch_bounds__(256) void frame_proj_wmma(
    const float* __restrict__ x, const int* __restrict__ ids,
    const float* __restrict__ W, const float* __restrict__ bia,
    float* __restrict__ out, int batch)
{
    // B fragments pre-swizzled for V_WMMA_F32_16X16X4_F32:
    // wfrag[((s*4 + nt)*32 + lane)*2 + j] = W_aug[4*s + (lane>=16?2:0) + j][nt*16 + (lane&15)]
    __shared__ float wfrag[16 * 4 * 32 * 2];                 // 16 KB
    __shared__ float astage[WAVES][ROWS_PER_WAVE * ROW_PAD]; // 34 KB

    const int tid  = threadIdx.x;
    const int lane = tid & 31;
    const int wave = tid >> 5;

    // ---- Build B-fragment table from W (4,64,14) and b (4,64) ----
    for (int f = tid; f < 16 * 4 * 32 * 2; f += 256) {
        int j  = f & 1;
        int ln = (f >> 1) & 31;
        int nt = (f >> 6) & 3;
        int s  = f >> 8;
        int k  = 4 * s + ((ln >= 16) ? 2 : 0) + j;
        int n  = nt * 16 + (ln & 15);
        float v = 0.0f;
        if (k < 56) {
            int e = k / 14, jj = k - 14 * e;
            v = W[(e * ND + n) * RD + jj];
        } else if (k < 60) {
            v = bia[(k - 56) * ND + n];
        }
        wfrag[f] = v;
    }
    __syncthreads();

    const long long rowBase = (long long)blockIdx.x * ROWS_PER_BLOCK
                            + (long long)wave * ROWS_PER_WAVE;

    // ---- Stage A tile (16 rows x 64 K) in LDS: 2 lanes cooperate per row ----
    const int m = lane & 15;                    // row within tile
    {
        long long row  = rowBase + m;
        long long rowc = (row < batch) ? row : (long long)(batch - 1);
        int e = ids[rowc];
        const float* xr = x + rowc * RD;        // 56B row stride -> 8B aligned pairs

        float* r0 = astage[wave] + m * ROW_PAD;
        // zero k=0..63 (split: lane<16 does first 8 float4s, lane>=16 the rest)
        v4f z = {0.f, 0.f, 0.f, 0.f};
        float* zb = r0 + ((lane >= 16) ? 32 : 0);
        #pragma unroll
        for (int i = 0; i < 8; ++i)
            *reinterpret_cast<v4f*>(zb + 4 * i) = z;
        asm volatile("" ::: "memory");          // zeros before scatter, program order
        // scatter x into expert slot: lane<16 writes pairs 0..3, lane>=16 pairs 4..6 + one-hot
        float* xs = r0 + 14 * e;
        if (lane < 16) {
            #pragma unroll
            for (int i = 0; i < 4; ++i)
                *reinterpret_cast<v2f*>(xs + 2 * i) =
                    *reinterpret_cast<const v2f*>(xr + 2 * i);
        } else {
            #pragma unroll
            for (int i = 4; i < 7; ++i)
                *reinterpret_cast<v2f*>(xs + 2 * i) =
                    *reinterpret_cast<const v2f*>(xr + 2 * i);
            r0[56 + e] = 1.0f;
        }
    }
    // Cross-lane LDS RAW within the wave: DS pipe is in-order per wave; drain + pin order.
    asm volatile("s_wait_dscnt 0" ::: "memory");

    // ---- 16x16x4 fp32 WMMA over K=64, N=64 (4 tiles of 16) ----
    v8f acc0 = {}, acc1 = {}, acc2 = {}, acc3 = {};
    const int khalf = (lane >= 16) ? 2 : 0;
    const float* arow = astage[wave] + m * ROW_PAD + khalf;

    #pragma unroll
    for (int s = 0; s < 16; ++s) {
        v2f a  = *reinterpret_cast<const v2f*>(arow + 4 * s);
        v2f b0 = *reinterpret_cast<const v2f*>(&wfrag[((s * 4 + 0) * 32 + lane) * 2]);
        v2f b1 = *reinterpret_cast<const v2f*>(&wfrag[((s * 4 + 1) * 32 + lane) * 2]);
        v2f b2 = *reinterpret_cast<const v2f*>(&wfrag[((s * 4 + 2) * 32 + lane) * 2]);
        v2f b3 = *reinterpret_cast<const v2f*>(&wfrag[((s * 4 + 3) * 32 + lane) * 2]);
        acc0 = __builtin_amdgcn_wmma_f32_16x16x4_f32(false, a, false, b0, (short)0, acc0, false, false);
        acc1 = __builtin_amdgcn_wmma_f32_16x16x4_f32(false, a, false, b1, (short)0, acc1, false, false);
        acc2 = __builtin_amdgcn_wmma_f32_16x16x4_f32(false, a, false, b2, (short)0, acc2, false, false);
        acc3 = __builtin_amdgcn_wmma_f32_16x16x4_f32(false, a, false, b3, (short)0, acc3, false, false);
    }

    // ---- Store: reg r -> row (r or r+8), col = nt*16 + (lane&15); coalesced b32 ----
    const int moff = (lane >= 16) ? 8 : 0;
    const int n0   = lane & 15;
    #pragma unroll
    for (int r = 0; r < 8; ++r) {
        long long orow = rowBase + r + moff;
        if (orow < batch) {
            float* op = out + orow * ND + n0;
            op[0]  = acc0[r];
            op[16] = acc1[r];
            op[32] = acc2[r];
            op[48] = acc3[r];
        }
    }
}

extern "C" void kernel_launch(void* const* d_in, const int* in_sizes, int n_in,
                              void* d_out, int out_size, void* d_ws, size_t ws_size,
                              hipStream_t stream) {
    const float* x   = (const float*)d_in[0];   // (batch, 14) fp32
    const int*   ids = (const int*)d_in[1];     // (batch,) int32
    const float* W   = (const float*)d_in[2];   // (4, 64, 14) fp32
    const float* bia = (const float*)d_in[3];   // (4, 64) fp32
    float* out = (float*)d_out;                 // (batch, 64) fp32

    int batch  = in_sizes[0] / RD;
    int blocks = (batch + ROWS_PER_BLOCK - 1) / ROWS_PER_BLOCK;
    frame_proj_wmma<<<blocks, 256, 0, stream>>>(x, ids, W, bia, out, batch);
}